// Baseline_65807488909790
// MI455X (gfx1250) — compile-verified
//
#include <hip/hip_runtime.h>
#include <hip/hip_bf16.h>

#define RES 8
#define FDIM (RES * RES * RES)   // 512
#define NCLS 40

typedef __attribute__((ext_vector_type(2))) float v2f;
typedef __attribute__((ext_vector_type(8))) float v8f;

// Monotone float<->uint encoding: unsigned comparisons on enc() match float order.
__device__ __forceinline__ unsigned enc_f32(float f) {
    unsigned u = __float_as_uint(f);
    return (u & 0x80000000u) ? ~u : (u | 0x80000000u);
}
__device__ __forceinline__ float dec_f32(unsigned e) {
    unsigned u = (e & 0x80000000u) ? (e ^ 0x80000000u) : ~e;
    return __uint_as_float(u);
}

// ---------------------------------------------------------------- init scratch
__global__ void init_ws(unsigned* __restrict__ counts,
                        unsigned* __restrict__ minmax, int B) {
    int i = blockIdx.x * blockDim.x + threadIdx.x;
    if (i < B * FDIM) counts[i] = 0u;
    if (i < B * 6)    minmax[i] = ((i % 6) < 3) ? 0xFFFFFFFFu : 0u; // min=+NaN, max=-NaN
}

// ------------------------------------------------------------- pass 1: min/max
__global__ void minmax_kernel(const float* __restrict__ x,
                              unsigned* __restrict__ minmax, int N) {
    const int b = blockIdx.y;
    const float* xb = x + (size_t)b * N * 3;

    float mn[3] = { INFINITY,  INFINITY,  INFINITY};
    float mx[3] = {-INFINITY, -INFINITY, -INFINITY};

    const int stride = gridDim.x * blockDim.x;
    for (int p = blockIdx.x * blockDim.x + threadIdx.x; p < N; p += stride) {
        const float* px = xb + 3 * (size_t)p;
#pragma unroll
        for (int d = 0; d < 3; ++d) {
            float v = px[d];
            mn[d] = fminf(mn[d], v);
            mx[d] = fmaxf(mx[d], v);
        }
    }

    // wave32 reduction
#pragma unroll
    for (int off = 16; off >= 1; off >>= 1) {
#pragma unroll
        for (int d = 0; d < 3; ++d) {
            mn[d] = fminf(mn[d], __shfl_xor(mn[d], off, 32));
            mx[d] = fmaxf(mx[d], __shfl_xor(mx[d], off, 32));
        }
    }

    __shared__ float smn[8 * 3], smx[8 * 3];
    const int wid  = threadIdx.x >> 5;
    const int lane = threadIdx.x & 31;
    if (lane == 0) {
#pragma unroll
        for (int d = 0; d < 3; ++d) { smn[wid * 3 + d] = mn[d]; smx[wid * 3 + d] = mx[d]; }
    }
    __syncthreads();
    if (threadIdx.x == 0) {
        const int nw = blockDim.x >> 5;
        for (int w = 1; w < nw; ++w)
#pragma unroll
            for (int d = 0; d < 3; ++d) {
                smn[d] = fminf(smn[d], smn[w * 3 + d]);
                smx[d] = fmaxf(smx[d], smx[w * 3 + d]);
            }
#pragma unroll
        for (int d = 0; d < 3; ++d) {
            atomicMin(&minmax[b * 6 + d],     enc_f32(smn[d]));
            atomicMax(&minmax[b * 6 + 3 + d], enc_f32(smx[d]));
        }
    }
}

// ----------------------------------------------------------- pass 2: histogram
__global__ void hist_kernel(const float* __restrict__ x,
                            const unsigned* __restrict__ minmax,
                            unsigned* __restrict__ counts, int N) {
    const int b = blockIdx.y;
    __shared__ unsigned h[FDIM];
    for (int i = threadIdx.x; i < FDIM; i += blockDim.x) h[i] = 0u;
    __syncthreads();

    float mn[3], scale[3];
#pragma unroll
    for (int d = 0; d < 3; ++d) {
        mn[d]      = dec_f32(minmax[b * 6 + d]);
        float mxv  = dec_f32(minmax[b * 6 + 3 + d]);
        float rng  = mxv - mn[d];
        scale[d]   = (float)RES / (rng > 0.f ? rng : 1.f);
    }

    const float* xb = x + (size_t)b * N * 3;
    const int stride = gridDim.x * blockDim.x;
    for (int p = blockIdx.x * blockDim.x + threadIdx.x; p < N; p += stride) {
        const float* px = xb + 3 * (size_t)p;
        int id[3];
#pragma unroll
        for (int d = 0; d < 3; ++d) {
            int v = (int)floorf((px[d] - mn[d]) * scale[d]);
            id[d] = v < 0 ? 0 : (v > RES - 1 ? RES - 1 : v);
        }
        int flat = (id[0] * RES + id[1]) * RES + id[2];
        atomicAdd(&h[flat], 1u);
    }
    __syncthreads();

    unsigned* cb = counts + (size_t)b * FDIM;
    for (int i = threadIdx.x; i < FDIM; i += blockDim.x) {
        unsigned v = h[i];
        if (v) atomicAdd(&cb[i], v);
    }
}

// -------------------------------------------- pass 3: (B x 512)@(512 x 40)+b
// One wave per 16x16 output tile. grid.x = ceil(B/16); blockDim = 32 * 3 waves.
// f32 WMMA 16x16x4 VGPR layout (ISA 7.12.2):
//   A 16x4 : lane&15 = M; VGPR0 holds K = (lane<16 ? 0 : 2), VGPR1 holds K+1
//   B 4x16 : lane&15 = N; VGPR0 holds K = (lane<16 ? 0 : 2), VGPR1 holds K+1
//   C/D    : VGPR i -> (M = i + (lane<16 ? 0 : 8), N = lane&15)
//
// Out-of-range m/n lanes are pointer-clamped to row/col 0 and loaded
// unconditionally: D[m][n] depends only on A-row m / B-col n, so garbage
// rows/cols only affect outputs we never store. No EXEC divergence.
__global__ void gemm_wmma(const unsigned* __restrict__ counts,
                          const float* __restrict__ W,
                          const float* __restrict__ bias,
                          float* __restrict__ out, int B, float invN) {
    const int lane  = threadIdx.x & 31;
    const int wid   = threadIdx.x >> 5;     // ntile 0..2
    const int mtile = blockIdx.x;
    const int ntile = wid;

    const int mrow  = mtile * 16 + (lane & 15);
    const int ncol  = ntile * 16 + (lane & 15);
    const int khalf = (lane >> 4) << 1;     // 0 or 2

    const bool nv = (ncol < NCLS);
    const unsigned* ap = counts + (size_t)(mrow < B ? mrow : 0) * FDIM + khalf;
    const float*    bp = W      + (size_t)(nv ? ncol : 0) * FDIM + khalf;

    v8f acc = {};
#pragma unroll 8
    for (int k0 = 0; k0 < FDIM; k0 += 4) {
        uint2  av = *(const uint2*)(ap + k0);   // global_load_b64
        float2 bv = *(const float2*)(bp + k0);  // global_load_b64
        v2f a, bm;
        a.x  = (float)av.x * invN;
        a.y  = (float)av.y * invN;
        bm.x = bv.x;
        bm.y = bv.y;
        acc = __builtin_amdgcn_wmma_f32_16x16x4_f32(
            /*neg_a=*/false, a, /*neg_b=*/false, bm,
            /*c_mod=*/(short)0, acc, /*reuse_a=*/false, /*reuse_b=*/false);
    }

    const float bv = nv ? bias[ncol] : 0.f;
#pragma unroll
    for (int i = 0; i < 8; ++i) {
        int m = mtile * 16 + i + ((lane >> 4) << 3);
        if (m < B && nv) out[(size_t)m * NCLS + ncol] = acc[i] + bv;
    }
}

// ---------------------------------------------------------------------- launch
extern "C" void kernel_launch(void* const* d_in, const int* in_sizes, int n_in,
                              void* d_out, int out_size, void* d_ws, size_t ws_size,
                              hipStream_t stream) {
    const float* x    = (const float*)d_in[0];  // (B, N, 3)
    const float* W    = (const float*)d_in[1];  // (40, 512)
    const float* bias = (const float*)d_in[2];  // (40,)
    float* out        = (float*)d_out;          // (B, 40)

    const int B = out_size / NCLS;
    const int N = (int)((long long)in_sizes[0] / (3LL * B));

    unsigned* counts = (unsigned*)d_ws;                    // B*512 u32
    unsigned* minmax = counts + (size_t)B * FDIM;          // B*6   u32 (enc min[3], max[3])

    init_ws<<<(B * FDIM + 255) / 256, 256, 0, stream>>>(counts, minmax, B);

    dim3 g1(32, B);
    minmax_kernel<<<g1, 256, 0, stream>>>(x, minmax, N);

    dim3 g2(64, B);
    hist_kernel<<<g2, 256, 0, stream>>>(x, minmax, counts, N);

    const int ntiles = (NCLS + 15) / 16;                   // 3
    gemm_wmma<<<(B + 15) / 16, 32 * ntiles, 0, stream>>>(counts, W, bias, out, B,
                                                         1.0f / (float)N);
}